// ParametricInterpolation_5961414606875
// MI455X (gfx1250) — compile-verified
//
#include <hip/hip_runtime.h>
#include <stdint.h>

#define SIG_LEN 2048
#define BLOCK   256
#define PER_THREAD (SIG_LEN / BLOCK)   // 8 outputs per thread

typedef float v4f __attribute__((ext_vector_type(4)));

__global__ __launch_bounds__(BLOCK)
void parametric_interp_kernel(const float* __restrict__ x,
                              const float* __restrict__ params,
                              float* __restrict__ out)
{
    __shared__ float row[SIG_LEN];     // 8 KB staging for this batch row

    const int b   = blockIdx.x;
    const int tid = threadIdx.x;

    const float* xrow = x + (size_t)b * SIG_LEN;

    // ---- CDNA5 async DMA: global -> LDS, 16B/lane per op; 2 ops move the 8KB row.
    // Explicit address registers for both halves (no reliance on IOFFSET
    // touching the LDS side of the transfer).
    {
        unsigned lbase = (unsigned)(uintptr_t)(&row[0]);   // LDS byte offset
        unsigned goff0 = (unsigned)tid * 16u;              // byte offset in row
        unsigned goff1 = goff0 + 4096u;
        unsigned loff0 = lbase + goff0;
        unsigned loff1 = lbase + goff1;
        asm volatile(
            "global_load_async_to_lds_b128 %0, %2, %4\n\t"
            "global_load_async_to_lds_b128 %1, %3, %4"
            :
            : "v"(loff0), "v"(loff1), "v"(goff0), "v"(goff1), "s"(xrow)
            : "memory");
    }

    // Block-uniform coefficient fetch (scalar path) while the DMA is in flight.
    const float* pr = params + (size_t)b * 5;
    float p0 = pr[0] / 1.0e12f;
    float p1 = pr[1] / 1.0e8f;
    float p2 = pr[2] / 1.0e4f;
    float p3 = pr[3];
    float p4 = pr[4] / 10.0f;

    // Wave-local DMA completion, then whole-workgroup visibility.
    asm volatile("s_wait_asynccnt 0" ::: "memory");
    __syncthreads();

    float res[PER_THREAD];
    const int n0 = tid * PER_THREAD;

#pragma unroll
    for (int j = 0; j < PER_THREAD; ++j) {
        float i  = (float)(n0 + j);
        float i2 = i * i;
        float i3 = i2 * i;
        float i4 = i2 * i2;

        float curve = p0 * i4;
        curve = fmaf(p1, i3, curve);
        curve = fmaf(p2, i2, curve);
        curve = fmaf(p3, i,  curve);
        curve += p4;

        float ci = rintf(curve);                 // round-half-to-even == jnp.round
        float k  = curve - ci;                   // fractional part

        float posf = fminf(fmaxf(i - ci, 1.0f), (float)(SIG_LEN - 1));
        int   pos  = (int)posf;

        float x1 = row[pos];                     // adjacent pair -> ds_load_2addr
        float x2 = row[pos - 1];

        res[j] = x1 * (1.0f - k) + x2 * k;       // exact reference lerp form
    }

    // Streaming (non-temporal) coalesced stores: 32B contiguous per thread.
    v4f* outv = (v4f*)(out + (size_t)b * SIG_LEN + n0);
    v4f lo = { res[0], res[1], res[2], res[3] };
    v4f hi = { res[4], res[5], res[6], res[7] };
    __builtin_nontemporal_store(lo, outv);
    __builtin_nontemporal_store(hi, outv + 1);
}

extern "C" void kernel_launch(void* const* d_in, const int* in_sizes, int n_in,
                              void* d_out, int out_size, void* d_ws, size_t ws_size,
                              hipStream_t stream)
{
    const float* x      = (const float*)d_in[0];   // [B, 2048] f32
    const float* params = (const float*)d_in[1];   // [B, 5]    f32
    float*       out    = (float*)d_out;           // [B, 2048] f32

    const int batch = in_sizes[1] / 5;             // 16384
    parametric_interp_kernel<<<batch, BLOCK, 0, stream>>>(x, params, out);
}